// EstimatorQNN__gen384_65481071406129
// MI455X (gfx1250) — compile-verified
//
#include <hip/hip_runtime.h>
#include <stdint.h>

typedef float v2f __attribute__((ext_vector_type(2)));
typedef float v8f __attribute__((ext_vector_type(8)));

#define NQ      18
#define BATCH   16
#define NSTATE  (1 << NQ)     // 262144 amplitudes per batch item
#define NPASS   20            // 5 group-gates x (1 RX layer + 3 RY layers)

// CDNA5 async global<->LDS paths (inline asm). Load path was accepted by the
// assembler in round 2; store path is symmetric. Flip to 0 on rejection.
#define TRY_ASYNC_LOAD  1
#define TRY_ASYNC_STORE 1

// ---------------------------------------------------------------------------
// Bit utilities: 18-bit state index; a "group" is 4 bit positions (p0<p1<p2<p3).
// insert4 deposits a value into the non-group positions (ascending order).
// With compile-time positions these fold to a handful of shifts/ors.
// ---------------------------------------------------------------------------
__device__ __forceinline__ unsigned insert4(unsigned c, int p0, int p1, int p2, int p3) {
  unsigned idx = c;
  idx = ((idx >> p0) << (p0 + 1)) | (idx & ((1u << p0) - 1u));
  idx = ((idx >> p1) << (p1 + 1)) | (idx & ((1u << p1) - 1u));
  idx = ((idx >> p2) << (p2 + 1)) | (idx & ((1u << p2) - 1u));
  idx = ((idx >> p3) << (p3 + 1)) | (idx & ((1u << p3) - 1u));
  return idx;
}

__device__ __forceinline__ unsigned gbits(int g, int p0, int p1, int p2, int p3) {
  return ((unsigned)(g & 1) << p0) | ((unsigned)((g >> 1) & 1) << p1)
       | ((unsigned)((g >> 2) & 1) << p2) | ((unsigned)((g >> 3) & 1) << p3);
}

// Linear tile index k (0..255, address-ordered) <-> (group row g, column n).
__device__ __forceinline__ void kToGN(int gk, unsigned k, int& g, int& n) {
  if (gk == 0)      { g = (int)(k & 15u); n = (int)(k >> 4); }
  else if (gk == 4) { g = (int)((k & 3u) | (((k >> 6) & 3u) << 2)); n = (int)((k >> 2) & 15u); }
  else              { n = (int)(k & 15u); g = (int)(k >> 4); }
}

__device__ __forceinline__ unsigned kOfGN(int gk, int g, int n) {
  if (gk == 0)      return (unsigned)(g | (n << 4));
  else if (gk == 4) return (unsigned)((g & 3) | (n << 2) | (((g >> 2) & 3) << 6));
  else              return (unsigned)(n | (g << 4));
}

// Padded LDS index: +4 floats per 16 -> bank-conflict-free, keeps 16B chunks
// contiguous and aligned.
__device__ __forceinline__ unsigned padIdx(unsigned k) { return k + ((k >> 4) << 2); }

// ---------------------------------------------------------------------------
// Build the 16x16 grouped-gate matrices (complex for RX, real for RY).
// mats layout: [b][pass][2][256] floats (Gr then Gi).
// ---------------------------------------------------------------------------
__global__ __launch_bounds__(256)
void qsim_prep_mats(const float* __restrict__ x, const float* __restrict__ params,
                    float* __restrict__ mats) {
  int t = blockIdx.x * 256 + threadIdx.x;
  if (t >= BATCH * NPASS * 256) return;
  int rc   = t & 255;
  int bp   = t >> 8;           // b*NPASS + pass
  int pass = bp % NPASS;
  int b    = bp / NPASS;
  int r = rc >> 4, c = rc & 15;
  int gk = pass % 5, layer = pass / 5;   // layer 0 = RX, 1..3 = RY layers 0..2

  float gr = 1.0f, gi = 0.0f;
  #pragma unroll
  for (int j = 0; j < 4; ++j) {
    int pos = (gk < 4) ? (4 * gk + j) : ((j == 0) ? 0 : (j == 1) ? 1 : (j == 2) ? 16 : 17);
    bool ident = (gk == 4) && (j < 2);
    int rb = (r >> j) & 1, cb = (c >> j) & 1;
    float er, ei;
    if (ident) {
      er = (rb == cb) ? 1.0f : 0.0f; ei = 0.0f;
    } else {
      int q = 17 - pos;
      float theta = (layer == 0) ? x[b * NQ + q] : params[NQ + (layer - 1) * NQ + q];
      float ch = cosf(0.5f * theta), sh = sinf(0.5f * theta);
      if (layer == 0) {               // RX: [[c, -i s], [-i s, c]]
        er = (rb == cb) ? ch : 0.0f;
        ei = (rb != cb) ? -sh : 0.0f;
      } else {                        // RY: [[c, -s], [s, c]]
        er = (rb == cb) ? ch : ((rb == 1 && cb == 0) ? sh : -sh);
        ei = 0.0f;
      }
    }
    float nr = gr * er - gi * ei;
    float ni = gr * ei + gi * er;
    gr = nr; gi = ni;
  }
  mats[(size_t)bp * 512 + rc]       = gr;   // Gr
  mats[(size_t)bp * 512 + 256 + rc] = gi;   // Gi
}

__global__ __launch_bounds__(256)
void qsim_init_state(float* __restrict__ sR, float* __restrict__ sI) {
  unsigned t = blockIdx.x * 256u + threadIdx.x;
  sR[t] = ((t & (NSTATE - 1u)) == 0u) ? 1.0f : 0.0f;
  sI[t] = 0.0f;
}

// ---------------------------------------------------------------------------
// One grouped-gate pass, fully specialized at compile time (GK, COMPLEX, CZ):
//   stage-in : global_load_async_to_lds_b128 (coalesced 16B chunks, padded LDS)
//   compute  : D = G x S via V_WMMA_F32_16X16X4_F32 (fp32 matrix path)
//   stage-out: s_wait_dscnt 0 ; global_store_async_from_lds_b128 (in place)
// Tiles are disjoint across waves -> the in-place pass is hazard-free.
// ---------------------------------------------------------------------------
template<int GK, int COMPLEX, int CZ>
__global__ __launch_bounds__(256)
void qsim_pass(float* __restrict__ sR, float* __restrict__ sI,
               const float* __restrict__ mats, int pass) {
  constexpr int p0 = (GK < 4) ? (4 * GK + 0) : 0;
  constexpr int p1 = (GK < 4) ? (4 * GK + 1) : 1;
  constexpr int p2 = (GK < 4) ? (4 * GK + 2) : 16;
  constexpr int p3 = (GK < 4) ? (4 * GK + 3) : 17;

  __shared__ __align__(16) float lds[8][2][320];   // [wave][plane][padded 256]

  const int lane  = threadIdx.x & 31;
  const int wv    = threadIdx.x >> 5;
  const int half  = lane >> 4;
  const int ln    = lane & 15;
  const int waveId = (blockIdx.x << 3) | wv;
  const int b     = waveId >> 10;          // 1024 tiles per batch item
  const int tile  = waveId & 1023;

  const unsigned T = insert4((unsigned)tile << 4, p0, p1, p2, p3);
  const unsigned batchBase = (unsigned)b << NQ;

  // ---- per-lane 8 address-consecutive tile elements
  const unsigned k0 = (unsigned)lane * 8u;
  int g0, n0; kToGN(GK, k0, g0, n0);
  const unsigned off0 = gbits(g0, p0, p1, p2, p3) | insert4((unsigned)n0, p0, p1, p2, p3);
  const unsigned gidx = batchBase + T + off0;      // 8 consecutive floats
  const unsigned pk0  = padIdx(k0);                // 2 contiguous 16B LDS chunks

  unsigned long long aR = (unsigned long long)(uintptr_t)(sR + gidx);
  unsigned long long aI = (unsigned long long)(uintptr_t)(sI + gidx);
  unsigned ldsR = (unsigned)(uintptr_t)&lds[wv][0][pk0];
  unsigned ldsI = (unsigned)(uintptr_t)&lds[wv][1][pk0];

  // ---- stage-in
#if TRY_ASYNC_LOAD
  // IOFFSET advances both LDS and global addresses (ISA async-to-LDS semantics)
  asm volatile("global_load_async_to_lds_b128 %0, %1, off"           :: "v"(ldsR), "v"(aR) : "memory");
  asm volatile("global_load_async_to_lds_b128 %0, %1, off offset:16" :: "v"(ldsR), "v"(aR) : "memory");
  asm volatile("global_load_async_to_lds_b128 %0, %1, off"           :: "v"(ldsI), "v"(aI) : "memory");
  asm volatile("global_load_async_to_lds_b128 %0, %1, off offset:16" :: "v"(ldsI), "v"(aI) : "memory");
  asm volatile("s_wait_asynccnt 0x0" ::: "memory");
#else
  {
    float4 r0 = *(const float4*)(sR + gidx);
    float4 r1 = *(const float4*)(sR + gidx + 4);
    float4 i0 = *(const float4*)(sI + gidx);
    float4 i1 = *(const float4*)(sI + gidx + 4);
    *(float4*)&lds[wv][0][pk0]     = r0;
    *(float4*)&lds[wv][0][pk0 + 4] = r1;
    *(float4*)&lds[wv][1][pk0]     = i0;
    *(float4*)&lds[wv][1][pk0 + 4] = i1;
  }
#endif
  __builtin_amdgcn_wave_barrier();   // scheduling fence (per-wave LDS staging)

  // ---- gather WMMA B operands from LDS (optionally fusing the CZ diagonal)
  const unsigned insLn = insert4((unsigned)ln, p0, p1, p2, p3);
  float bRv[8], bIv[8];
  #pragma unroll
  for (int kb = 0; kb < 4; ++kb) {
    #pragma unroll
    for (int e = 0; e < 2; ++e) {
      int g = 4 * kb + 2 * half + e;
      unsigned pk = padIdx(kOfGN(GK, g, ln));
      float r  = lds[wv][0][pk];
      float im = lds[wv][1][pk];
      if (CZ) {
        unsigned loc = T | gbits(g, p0, p1, p2, p3) | insLn;
        float f = (__popc(loc & (loc >> 1)) & 1) ? -1.0f : 1.0f;
        r *= f; im *= f;
      }
      bRv[kb * 2 + e] = r;
      bIv[kb * 2 + e] = im;
    }
  }

  const float* Gr = mats + (size_t)(b * NPASS + pass) * 512;
  const float* Gi = Gr + 256;

  v8f accR = {0.f, 0.f, 0.f, 0.f, 0.f, 0.f, 0.f, 0.f};
  v8f accI = {0.f, 0.f, 0.f, 0.f, 0.f, 0.f, 0.f, 0.f};

  #pragma unroll
  for (int kb = 0; kb < 4; ++kb) {
    v2f a;  a.x  = Gr[ln * 16 + 4 * kb + 2 * half];
            a.y  = Gr[ln * 16 + 4 * kb + 2 * half + 1];
    v2f bR; bR.x = bRv[2 * kb]; bR.y = bRv[2 * kb + 1];
    v2f bI; bI.x = bIv[2 * kb]; bI.y = bIv[2 * kb + 1];
    accR = __builtin_amdgcn_wmma_f32_16x16x4_f32(false, a, false, bR, (short)0, accR, false, false);
    accI = __builtin_amdgcn_wmma_f32_16x16x4_f32(false, a, false, bI, (short)0, accI, false, false);
  }
  if (COMPLEX) {
    #pragma unroll
    for (int kb = 0; kb < 4; ++kb) {
      v2f a;   a.x   = Gi[ln * 16 + 4 * kb + 2 * half];
               a.y   = Gi[ln * 16 + 4 * kb + 2 * half + 1];
      v2f nbI; nbI.x = -bIv[2 * kb]; nbI.y = -bIv[2 * kb + 1];
      v2f bR;  bR.x  = bRv[2 * kb];  bR.y  = bRv[2 * kb + 1];
      accR = __builtin_amdgcn_wmma_f32_16x16x4_f32(false, a, false, nbI, (short)0, accR, false, false);
      accI = __builtin_amdgcn_wmma_f32_16x16x4_f32(false, a, false, bR,  (short)0, accI, false, false);
    }
  }
  __builtin_amdgcn_wave_barrier();

  // ---- scatter D back into padded LDS (LDS ops are in-order per wave)
  #pragma unroll
  for (int i = 0; i < 8; ++i) {
    int g = i + 8 * half;
    unsigned pk = padIdx(kOfGN(GK, g, ln));
    lds[wv][0][pk] = accR[i];
    lds[wv][1][pk] = accI[i];
  }
  __builtin_amdgcn_wave_barrier();

  // ---- stage-out (in place)
#if TRY_ASYNC_STORE
  // Async ops are NOT ordered with DS ops -> drain DScnt before LDS is read.
  asm volatile("s_wait_dscnt 0x0" ::: "memory");
  asm volatile("global_store_async_from_lds_b128 %0, %1, off"           :: "v"(aR), "v"(ldsR) : "memory");
  asm volatile("global_store_async_from_lds_b128 %0, %1, off offset:16" :: "v"(aR), "v"(ldsR) : "memory");
  asm volatile("global_store_async_from_lds_b128 %0, %1, off"           :: "v"(aI), "v"(ldsI) : "memory");
  asm volatile("global_store_async_from_lds_b128 %0, %1, off offset:16" :: "v"(aI), "v"(ldsI) : "memory");
  // S_ENDPGM performs an implicit wait-idle, completing the async stores
  // before the next stream-ordered pass can start.
#else
  {
    float4 r0 = *(const float4*)&lds[wv][0][pk0];
    float4 r1 = *(const float4*)&lds[wv][0][pk0 + 4];
    float4 i0 = *(const float4*)&lds[wv][1][pk0];
    float4 i1 = *(const float4*)&lds[wv][1][pk0 + 4];
    *(float4*)(sR + gidx)     = r0;
    *(float4*)(sR + gidx + 4) = r1;
    *(float4*)(sI + gidx)     = i0;
    *(float4*)(sI + gidx + 4) = i1;
  }
#endif
}

// ---------------------------------------------------------------------------
// Deterministic two-stage Z-expectation reduction.
// ---------------------------------------------------------------------------
__global__ __launch_bounds__(256)
void qsim_reduce1(const float* __restrict__ sR, const float* __restrict__ sI,
                  float* __restrict__ partials) {
  int b = blockIdx.x >> 6;
  int chunk = blockIdx.x & 63;
  int tid = threadIdx.x;
  unsigned base = (unsigned)b << NQ;

  float acc[NQ];
  #pragma unroll
  for (int q = 0; q < NQ; ++q) acc[q] = 0.0f;

  for (int k = 0; k < 16; ++k) {
    unsigned loc = (unsigned)chunk * 4096u + (unsigned)tid + (unsigned)k * 256u;
    float r  = sR[base + loc];
    float im = sI[base + loc];
    float p = r * r + im * im;
    #pragma unroll
    for (int q = 0; q < NQ; ++q)
      acc[q] += ((loc >> (17 - q)) & 1u) ? -p : p;
  }

  __shared__ float red[256];
  for (int q = 0; q < NQ; ++q) {
    red[tid] = acc[q];
    __syncthreads();
    for (int s = 128; s > 0; s >>= 1) {
      if (tid < s) red[tid] += red[tid + s];
      __syncthreads();
    }
    if (tid == 0) partials[(size_t)blockIdx.x * NQ + q] = red[0];
    __syncthreads();
  }
}

__global__ __launch_bounds__(512)
void qsim_reduce2(const float* __restrict__ partials, float* __restrict__ out) {
  int t = threadIdx.x;
  if (t < BATCH * NQ) {
    int b = t / NQ, q = t % NQ;
    float s = 0.0f;
    for (int k = 0; k < 64; ++k) s += partials[(size_t)(b * 64 + k) * NQ + q];
    out[t] = s;
  }
}

// ---------------------------------------------------------------------------
// Host orchestration.
// ---------------------------------------------------------------------------
static inline void launch_pass(int gk, int isComplex, int applyCZ, int pass,
                               float* sR, float* sI, const float* mats,
                               hipStream_t stream) {
  dim3 g(2048), blk(256);
  if (applyCZ) {                       // only occurs for gk==0, real layer
    qsim_pass<0, 0, 1><<<g, blk, 0, stream>>>(sR, sI, mats, pass);
    return;
  }
  if (isComplex) {
    switch (gk) {
      case 0: qsim_pass<0, 1, 0><<<g, blk, 0, stream>>>(sR, sI, mats, pass); break;
      case 1: qsim_pass<1, 1, 0><<<g, blk, 0, stream>>>(sR, sI, mats, pass); break;
      case 2: qsim_pass<2, 1, 0><<<g, blk, 0, stream>>>(sR, sI, mats, pass); break;
      case 3: qsim_pass<3, 1, 0><<<g, blk, 0, stream>>>(sR, sI, mats, pass); break;
      default: qsim_pass<4, 1, 0><<<g, blk, 0, stream>>>(sR, sI, mats, pass); break;
    }
  } else {
    switch (gk) {
      case 0: qsim_pass<0, 0, 0><<<g, blk, 0, stream>>>(sR, sI, mats, pass); break;
      case 1: qsim_pass<1, 0, 0><<<g, blk, 0, stream>>>(sR, sI, mats, pass); break;
      case 2: qsim_pass<2, 0, 0><<<g, blk, 0, stream>>>(sR, sI, mats, pass); break;
      case 3: qsim_pass<3, 0, 0><<<g, blk, 0, stream>>>(sR, sI, mats, pass); break;
      default: qsim_pass<4, 0, 0><<<g, blk, 0, stream>>>(sR, sI, mats, pass); break;
    }
  }
}

extern "C" void kernel_launch(void* const* d_in, const int* in_sizes, int n_in,
                              void* d_out, int out_size, void* d_ws, size_t ws_size,
                              hipStream_t stream) {
  (void)in_sizes; (void)n_in; (void)out_size; (void)ws_size;
  const float* x      = (const float*)d_in[0];   // (16, 18) f32
  const float* params = (const float*)d_in[1];   // (72,)   f32
  float* out = (float*)d_out;                    // (16, 18) f32
  float* ws  = (float*)d_ws;

  float* sR       = ws;                                   // 16 * 2^18 floats
  float* sI       = sR + (size_t)BATCH * NSTATE;          // 16 * 2^18 floats
  float* mats     = sI + (size_t)BATCH * NSTATE;          // 16 * 20 * 512 floats
  float* partials = mats + (size_t)BATCH * NPASS * 512;   // 1024 * 18 floats

  qsim_prep_mats<<<(BATCH * NPASS * 256 + 255) / 256, 256, 0, stream>>>(x, params, mats);
  qsim_init_state<<<(BATCH * NSTATE) / 256, 256, 0, stream>>>(sR, sI);

  for (int pass = 0; pass < NPASS; ++pass) {
    int gk = pass % 5;
    int layer = pass / 5;
    int isComplex = (layer == 0) ? 1 : 0;                 // RX layer is complex
    int applyCZ = ((pass == 10) || (pass == 15)) ? 1 : 0; // CZ after RY layers 0,1
    // (CZ after the final RY layer doesn't change |amp|^2 -> dropped)
    launch_pass(gk, isComplex, applyCZ, pass, sR, sI, mats, stream);
  }

  qsim_reduce1<<<BATCH * 64, 256, 0, stream>>>(sR, sI, partials);
  qsim_reduce2<<<1, 512, 0, stream>>>(partials, out);
}